// SpinalCPG_3246995276436
// MI455X (gfx1250) — compile-verified
//
#include <hip/hip_runtime.h>
#include <stdint.h>

// ---------------- problem constants (from the reference) ----------------
#define TT 512          // timesteps
#define BB 2048         // batch
#define NN 96           // neurons per batch element (2*NS)
#define CHUNK 8         // timesteps per TDM tile (3 KB per tile)
#define NCHUNK (TT / CHUNK)
#define WPB 8           // waves (= batch elements) per block
#define LPW 32          // wave32

#define TAU_M     0.5f
#define TAU_SYN   0.6f
#define V_TH      0.5f
#define NOISE_STD 0.15f
#define W_V2A_MN  0.7f
#define W_V2A_V0D 0.6f
#define W_V0D_X   0.9f
#define W_V2A_DI6 0.4f
#define W_DI6_V2A 0.3f
#define W_DI6_MN  0.2f
#define W_MN_RSH  0.5f
#define W_RSH_V2A 0.6f

typedef unsigned int v4u __attribute__((ext_vector_type(4)));
typedef int          v8i __attribute__((ext_vector_type(8)));
typedef int          v4i __attribute__((ext_vector_type(4)));

// Issue one TDM 2-D tile load: CHUNK rows x 96 f32, row stride = B*96 elems,
// global -> LDS (contiguous rows).  D# built per CDNA5 ISA.
// This toolchain exposes the 6-arg builtin form:
//   (uint32x4 g0, int32x8 g1, int32x4 g2, int32x4 g3, int32x8 extra, i32 cpol)
__device__ __forceinline__ void tdm_issue(const float* gsrc, unsigned lds_byte_off) {
  const uint64_t ga = (uint64_t)(uintptr_t)gsrc;
  v4u g0;
  g0[0] = 1u;                                               // count=1, user load D#
  g0[1] = lds_byte_off;                                     // lds_addr
  g0[2] = (unsigned)(ga & 0xffffffffu);                     // global_addr[31:0]
  g0[3] = (unsigned)((ga >> 32) & 0x01ffffffu) | (2u << 30); // global_addr[56:32], type=2
  v8i g1;
  g1[0] = (int)(2u << 16);        // workgroup_mask=0, data_size=2 (4 bytes)
  g1[1] = (int)(NN << 16);        // tensor_dim0 = 96 (bits 79:48, low half)
  g1[2] = (int)(CHUNK << 16);     // tensor_dim0 hi=0 | tensor_dim1 = CHUNK
  g1[3] = (int)(NN << 16);        // tensor_dim1 hi=0 | tile_dim0 = 96
  g1[4] = (int)CHUNK;             // tile_dim1 = CHUNK, tile_dim2 = 0
  g1[5] = (int)(BB * NN);         // tensor_dim0_stride = B*96 elements
  g1[6] = 0;                      // stride hi / dim1_stride lo
  g1[7] = 0;
  v4i g2 = {0, 0, 0, 0};          // 2-D tensor: groups 2/3 unused
  v4i g3 = {0, 0, 0, 0};
  v8i gx = {0, 0, 0, 0, 0, 0, 0, 0};  // extra descriptor words (unused, 2-D case)
  __builtin_amdgcn_tensor_load_to_lds(g0, g1, g2, g3, gx, 0);
}

__global__ __launch_bounds__(WPB * LPW) void cpg_scan_kernel(
    const float* __restrict__ v0,
    const float* __restrict__ syn0,
    const float* __restrict__ noise,
    const float* __restrict__ drive_in,
    const float* __restrict__ turn_in,
    const int*   __restrict__ glide0,
    const int*   __restrict__ bout0,
    const unsigned char* __restrict__ inbout0,
    float*       __restrict__ out)
{
  __shared__ float nbuf[WPB][2][CHUNK * NN];   // double-buffered noise tiles
  __shared__ float stage[WPB][NN];             // one-time syn0 staging

  const int lane = threadIdx.x & 31;
  const int w    = threadIdx.x >> 5;
  const int b    = blockIdx.x * WPB + w;       // one wave == one batch element

  // ---- per-neuron membrane state: lane owns neurons lane, lane+32, lane+64 ----
  const size_t sbase = (size_t)b * NN;
  float v_0 = v0[sbase + lane];
  float v_1 = v0[sbase + lane + 32];
  float v_2 = v0[sbase + lane + 64];

  // ---- fold syn0 into the 10 segment sums (replicated across lanes) ----
  stage[w][lane]      = syn0[sbase + lane];
  stage[w][lane + 32] = syn0[sbase + lane + 32];
  stage[w][lane + 64] = syn0[sbase + lane + 64];
  __syncthreads();
  float s[10];
  {
    const float* sp = stage[w];
    const int lo[10] = {0, 16, 24, 28, 40, 48, 64, 72, 76, 88};
    const int hi[10] = {16, 24, 28, 40, 48, 64, 72, 76, 88, 96};
#pragma unroll
    for (int g = 0; g < 10; ++g) {
      float a = 0.f;
      for (int i = lo[g]; i < hi[g]; ++i) a += sp[i];
      s[g] = a;
    }
  }

  // ---- wave-uniform scalars ----
  const float drive   = drive_in[b];
  const float turn    = turn_in[b];
  const float drive_L = drive + fmaxf(0.f, -turn) * 0.3f;
  const float drive_R = drive + fmaxf(0.f,  turn) * 0.3f;
  int bout_len  = (int)(drive * 3.0f);          if (bout_len  < 1) bout_len  = 1;
  int glide_len = (int)((1.0f - drive) * 3.0f); if (glide_len < 1) glide_len = 1;
  int  glide   = glide0[b];
  int  bout    = bout0[b];
  bool in_bout = inbout0[b] != 0;

  // lane->group selectors (compile-time segment map)
  const bool l8  = lane < 8,  l12 = lane < 12, l16 = lane < 16;
  const bool l24 = lane < 24, l28 = lane < 28;

  const unsigned lds0 = (unsigned)(uintptr_t)&nbuf[w][0][0];
  const unsigned lds1 = (unsigned)(uintptr_t)&nbuf[w][1][0];

  // prologue: prefetch chunk 0
  tdm_issue(noise + (size_t)b * NN, lds0);

  for (int c = 0; c < NCHUNK; ++c) {
    if (c + 1 < NCHUNK) {
      // WAR safety: all DS reads of the buffer we are about to overwrite drained
      asm volatile("s_wait_dscnt 0x0" ::: "memory");
      tdm_issue(noise + ((size_t)((c + 1) * CHUNK) * BB + b) * NN,
                (c & 1) ? lds0 : lds1);
      __builtin_amdgcn_s_wait_tensorcnt((short)1);  // chunk c resident
    } else {
      __builtin_amdgcn_s_wait_tensorcnt((short)0);
    }
    asm volatile("" ::: "memory");                  // keep DS reads below the wait
    const float* nb = nbuf[w][c & 1];

#pragma unroll
    for (int j = 0; j < CHUNK; ++j) {
      const int t = c * CHUNK + j;
      float4 o;
      if (__builtin_amdgcn_readfirstlane(glide) > 0) {   // uniform glide branch
        --glide;
        o = make_float4(0.f, 0.f, 0.f, 0.f);
      } else {
        // means of previous-step syn (group sums * 1/len)
        const float m0 = s[0] * (1.f/16.f), m1 = s[1] * (1.f/8.f);
        const float m2 = s[2] * (1.f/4.f),  m3 = s[3] * (1.f/12.f);
        const float m4 = s[4] * (1.f/8.f),  m5 = s[5] * (1.f/16.f);
        const float m6 = s[6] * (1.f/8.f),  m7 = s[7] * (1.f/4.f);
        const float m8 = s[8] * (1.f/12.f), m9 = s[9] * (1.f/8.f);

        const float iv2aL = drive_L*2.f + 0.5f - W_V0D_X*m6 - W_RSH_V2A*m4 - W_DI6_V2A*m2;
        const float iv0dL = W_V2A_V0D*m0;
        const float idi6L = W_V2A_DI6*m0;
        const float imnL  = W_V2A_MN*m0 - W_DI6_MN*m2;
        const float irshL = W_MN_RSH*m3;
        const float iv2aR = drive_R*2.f + 0.5f - W_V0D_X*m1 - W_RSH_V2A*m9 - W_DI6_V2A*m7;
        const float iv0dR = W_V2A_V0D*m5;
        const float idi6R = W_V2A_DI6*m5;
        const float imnR  = W_V2A_MN*m5 - W_DI6_MN*m7;
        const float irshR = W_MN_RSH*m8;

        // per-lane current pick (segments: 16|8|4|12|8 per side)
        const float I0 = l16 ? iv2aL : (l24 ? iv0dL : (l28 ? idi6L : imnL));
        const float I1 = l8  ? imnL  : (l16 ? irshL : iv2aR);
        const float I2 = l8  ? iv0dR : (l12 ? idi6R : (l24 ? imnR : irshR));

        const float n0 = nb[j * NN + lane];
        const float n1 = nb[j * NN + lane + 32];
        const float n2 = nb[j * NN + lane + 64];

        const float vn0 = TAU_M * v_0 + (1.f - TAU_M) * (I0 + n0 * NOISE_STD);
        const float vn1 = TAU_M * v_1 + (1.f - TAU_M) * (I1 + n1 * NOISE_STD);
        const float vn2 = TAU_M * v_2 + (1.f - TAU_M) * (I2 + n2 * NOISE_STD);

        const unsigned sp0 = __builtin_amdgcn_ballot_w32(vn0 >= V_TH);
        const unsigned sp1 = __builtin_amdgcn_ballot_w32(vn1 >= V_TH);
        const unsigned sp2 = __builtin_amdgcn_ballot_w32(vn2 >= V_TH);
        v_0 = (vn0 >= V_TH) ? 0.f : vn0;
        v_1 = (vn1 >= V_TH) ? 0.f : vn1;
        v_2 = (vn2 >= V_TH) ? 0.f : vn2;

        // syn group sums: S = tau*S + popcount(spikes in segment)  (SALU co-exec)
        s[0] = TAU_SYN*s[0] + (float)__builtin_popcount(sp0 & 0x0000FFFFu);
        s[1] = TAU_SYN*s[1] + (float)__builtin_popcount(sp0 & 0x00FF0000u);
        s[2] = TAU_SYN*s[2] + (float)__builtin_popcount(sp0 & 0x0F000000u);
        s[3] = TAU_SYN*s[3] + (float)(__builtin_popcount(sp0 & 0xF0000000u) +
                                      __builtin_popcount(sp1 & 0x000000FFu));
        s[4] = TAU_SYN*s[4] + (float)__builtin_popcount(sp1 & 0x0000FF00u);
        s[5] = TAU_SYN*s[5] + (float)__builtin_popcount(sp1 & 0xFFFF0000u);
        s[6] = TAU_SYN*s[6] + (float)__builtin_popcount(sp2 & 0x000000FFu);
        s[7] = TAU_SYN*s[7] + (float)__builtin_popcount(sp2 & 0x00000F00u);
        s[8] = TAU_SYN*s[8] + (float)__builtin_popcount(sp2 & 0x00FFF000u);
        s[9] = TAU_SYN*s[9] + (float)__builtin_popcount(sp2 & 0xFF000000u);

        // bout / glide state machine (uniform, replicated)
        const float mL = s[3] * (1.f/12.f), mR = s[8] * (1.f/12.f);
        const bool start = (!in_bout) && (fmaxf(mL, mR) > 0.15f);
        if (start) { bout = bout_len; in_bout = true; }
        if (in_bout) --bout;
        if (in_bout && bout <= 0) { in_bout = false; glide = glide_len; }
        o = make_float4(mL, mR, (mL + mR) * 0.5f, (mR - mL) * 2.f);
      }
      if (lane == 0) {
        *reinterpret_cast<float4*>(out + ((size_t)t * BB + b) * 4) = o;
      }
    }
  }
}

extern "C" void kernel_launch(void* const* d_in, const int* in_sizes, int n_in,
                              void* d_out, int out_size, void* d_ws, size_t ws_size,
                              hipStream_t stream) {
  (void)in_sizes; (void)n_in; (void)out_size; (void)d_ws; (void)ws_size;
  const float* v0     = (const float*)d_in[0];
  const float* syn0   = (const float*)d_in[1];
  const float* noise  = (const float*)d_in[2];
  const float* drive  = (const float*)d_in[3];
  const float* turn   = (const float*)d_in[4];
  const int*   glide0 = (const int*)d_in[5];
  const int*   bout0  = (const int*)d_in[6];
  const unsigned char* inb0 = (const unsigned char*)d_in[7];
  float* out = (float*)d_out;

  dim3 grid(BB / WPB);
  dim3 block(WPB * LPW);
  hipLaunchKernelGGL(cpg_scan_kernel, grid, block, 0, stream,
                     v0, syn0, noise, drive, turn, glide0, bout0, inb0, out);
}